// HiPAttention_86818468921805
// MI455X (gfx1250) — compile-verified
//
#include <hip/hip_runtime.h>
#include <hip/hip_bf16.h>
#include <math.h>

// ---------------------------------------------------------------------------
// HiP-attention (hierarchical pruning attention), fp32, MI455X / gfx1250.
//   B=2, S=2048, E=1024, H=16, D=64, CHUNK=32, TOPK=8 -> 16 final keys/query.
// Pipeline:
//   1) gemm_f32_wmma (x3): Q/K/V = x@W + b, written in [B,H,S,D] layout
//   2) hip_prune_attn: hierarchical top-k pruning + 16-key sparse attention
//   3) gemm_f32_wmma: out = ctx@Wo + bo
// GEMM: V_WMMA_F32_16X16X4_F32. W tile double-buffered in LDS via scatter
// GLOBAL_LOAD_ASYNC_TO_LDS_B32 in a K-pair-interleaved layout so each B
// fragment is a single aligned ds_load_b64 (no repacking movs).
// ---------------------------------------------------------------------------

typedef __attribute__((ext_vector_type(2))) float v2f;
typedef __attribute__((ext_vector_type(8))) float v8f;

#define E_DIM 1024
#define S_LEN 2048
#define H_NUM 16
#define D_HEAD 64
#define KC 32               // K-chunk staged in LDS per double-buffer slot

// LDS layout for one chunk: pair p = k>>1 (16 pairs), col c (64):
//   shW[buf][(p*64 + c)*2 + (k&1)]  ->  fragment {W[2p][c], W[2p+1][c]} is
//   8 contiguous bytes -> one ds_load_b64.
#define CHUNK_FLOATS (KC * 64)          // 2048 floats = 8 KB per buffer

// ---------------------------------------------------------------------------
// fp32 WMMA GEMM:  out[4096 x 1024] = A[4096 x 1024] @ W[1024 x 1024] + bias
// Block: 256 threads = 8 waves; block tile 128(M) x 64(N); wave tile 16 x 64.
// A-frag (16x4 f32, ISA 7.12.2): lane(0..15)=M, VGPRs = K{0,1}; hi lanes K{2,3}.
// B-frag (4x16 f32, C-matrix analogy): lane%16=N, VGPR i = K{i} (+2 hi half).
// ---------------------------------------------------------------------------
__global__ __launch_bounds__(256) void gemm_f32_wmma(
    const float* __restrict__ A, const float* __restrict__ W,
    const float* __restrict__ bias, float* __restrict__ out,
    int qkv_layout)
{
    __shared__ float shW[2][CHUNK_FLOATS];         // 16 KB double buffer

    const int wave  = threadIdx.x >> 5;
    const int lane  = threadIdx.x & 31;
    const int lmod  = lane & 15;
    const int lhalf = lane >> 4;

    const int m0 = blockIdx.x * 128 + wave * 16;   // 4096 / 128 = 32 blocks x
    const int n0 = blockIdx.y * 64;                // 1024 / 64  = 16 blocks y

    const float* __restrict__ arow = A + (size_t)(m0 + lmod) * E_DIM + lhalf * 2;

    // Cooperative scatter-stager: thread handles 8 elements per chunk.
    // element idx e (0..2047): k = e>>6 (row in chunk), c = e&63 (col).
    unsigned lds_off[8];      // LDS byte offsets within a buffer
    unsigned gbl_off[8];      // global float offsets within a chunk
#pragma unroll
    for (int i = 0; i < 8; ++i) {
        const int e = threadIdx.x + i * 256;
        const int k = e >> 6;
        const int c = e & 63;
        lds_off[i] = (unsigned)((((k >> 1) * 64 + c) * 2 + (k & 1)) * 4);
        gbl_off[i] = (unsigned)(k * E_DIM + c);
    }
    const float* __restrict__ wsrc = W + n0;       // + (chunk*KC)*E_DIM
    const unsigned shW_base0 = (unsigned)(uintptr_t)&shW[0][0];
    const unsigned shW_base1 = (unsigned)(uintptr_t)&shW[1][0];

    v8f acc[4];
    v8f zero = {0.f, 0.f, 0.f, 0.f, 0.f, 0.f, 0.f, 0.f};
#pragma unroll
    for (int t = 0; t < 4; ++t) acc[t] = zero;

    // Prologue: async-stage chunk 0 into buffer 0 (ASYNCcnt path).
#pragma unroll
    for (int i = 0; i < 8; ++i) {
        unsigned dst = shW_base0 + lds_off[i];
        unsigned long long ga =
            (unsigned long long)(uintptr_t)(wsrc + gbl_off[i]);
        asm volatile("global_load_async_to_lds_b32 %0, %1, off"
                     :: "v"(dst), "v"(ga) : "memory");
    }

    const int NC = E_DIM / KC;                     // 32 chunks
    for (int c = 0; c < NC; ++c) {
        if (c + 1 < NC) {
            // Stage next chunk into the other buffer, then wait until only
            // those 8 new ops are outstanding (async loads retire in order
            // => chunk c has fully landed for this wave).
            const unsigned base = ((c + 1) & 1) ? shW_base1 : shW_base0;
            const float* src = wsrc + (size_t)(c + 1) * KC * E_DIM;
#pragma unroll
            for (int i = 0; i < 8; ++i) {
                unsigned dst = base + lds_off[i];
                unsigned long long ga =
                    (unsigned long long)(uintptr_t)(src + gbl_off[i]);
                asm volatile("global_load_async_to_lds_b32 %0, %1, off"
                             :: "v"(dst), "v"(ga) : "memory");
            }
            asm volatile("s_wait_asynccnt 8" ::: "memory");
        } else {
            asm volatile("s_wait_asynccnt 0" ::: "memory");
        }
        __syncthreads();   // every wave has waited -> whole tile visible

        const float* wl = &shW[c & 1][0];
        __builtin_prefetch(arow + (size_t)(c + 1) * KC, 0, 3);  // A stream

#pragma unroll
        for (int kk = 0; kk < KC; kk += 4) {
            v2f a = *(const v2f*)(arow + c * KC + kk);
            // pair index = (kk + 2*lhalf)/2 ; fragment = 8 contiguous bytes
            const v2f* wp = (const v2f*)(wl + ((kk >> 1) + lhalf) * 128 + lmod * 2);

            v2f b0 = wp[0];        // cols n0 + 0..15
            v2f b1 = wp[16];       // cols n0 + 16..31
            v2f b2 = wp[32];       // cols n0 + 32..47
            v2f b3 = wp[48];       // cols n0 + 48..63

            acc[0] = __builtin_amdgcn_wmma_f32_16x16x4_f32(
                false, a, false, b0, (short)0, acc[0], false, false);
            acc[1] = __builtin_amdgcn_wmma_f32_16x16x4_f32(
                false, a, false, b1, (short)0, acc[1], false, false);
            acc[2] = __builtin_amdgcn_wmma_f32_16x16x4_f32(
                false, a, false, b2, (short)0, acc[2], false, false);
            acc[3] = __builtin_amdgcn_wmma_f32_16x16x4_f32(
                false, a, false, b3, (short)0, acc[3], false, false);
        }
        __syncthreads();   // done reading before the buffer is re-targeted
    }

    // C/D layout: VGPR i, lanes 0-15 -> (M=i, N=lane); lanes 16-31 -> (M=i+8).
#pragma unroll
    for (int i = 0; i < 8; ++i) {
        const int row = m0 + i + 8 * lhalf;
#pragma unroll
        for (int t = 0; t < 4; ++t) {
            const int col = n0 + t * 16 + lmod;
            const float v = acc[t][i] + bias[col];
            if (qkv_layout) {
                const int b = row >> 11;          // row = b*S + s
                const int s = row & (S_LEN - 1);
                const int h = col >> 6;           // col = h*64 + d
                const int d = col & (D_HEAD - 1);
                out[(((size_t)b * H_NUM + h) * S_LEN + s) * D_HEAD + d] = v;
            } else {
                out[(size_t)row * E_DIM + col] = v;
            }
        }
    }
}

// ---------------------------------------------------------------------------
// Hierarchical pruning + 16-key sparse attention. One lane == one query.
// Block = 128 threads (4 waves); a block covers 128 consecutive s of one (b,h).
// Surviving index set is always 8 contiguous blocks of size bs (32->16->8->4->2)
// so we only track 8 base indices per query.
// ---------------------------------------------------------------------------
#define PB 128  // threads per block

__global__ __launch_bounds__(PB) void hip_prune_attn(
    const float* __restrict__ Q, const float* __restrict__ K,
    const float* __restrict__ V, float* __restrict__ CTX)
{
    __shared__ float ldsk[64 * D_HEAD];   // level-0 rep K rows (16 KB)
    __shared__ float scl0[64 * PB];       // per-lane level-0 scores (32 KB)
    __shared__ float scl1[16 * PB];       // per-lane level scores   (8 KB)
    __shared__ int   basel[8 * PB];       // per-lane block bases    (4 KB)

    const int tid = threadIdx.x;
    const int qid = blockIdx.x * PB + tid;       // ((b*H+h)*S + s)
    const int bh  = qid >> 11;                   // same for whole block
    const int s   = qid & (S_LEN - 1);

    const float* __restrict__ kbase = K + (size_t)bh * S_LEN * D_HEAD;
    const float* __restrict__ vbase = V + (size_t)bh * S_LEN * D_HEAD;

    // ---- stage level-0 representative K rows (rep = c*32+16, c=0..63) ----
#pragma unroll
    for (int i = 0; i < 8; ++i) {
        const int idx4 = tid + i * PB;           // float4 index, 0..1023
        const int c    = idx4 >> 4;              // 16 float4 per row
        const int d4   = idx4 & 15;
        const float4 val =
            ((const float4*)(kbase + (size_t)(c * 32 + 16) * D_HEAD))[d4];
        ((float4*)ldsk)[idx4] = val;
    }
    __syncthreads();

    // ---- load q into registers ----
    const float* __restrict__ qrow = Q + (size_t)qid * D_HEAD;
    float q[D_HEAD];
#pragma unroll
    for (int i = 0; i < 16; ++i) {
        const float4 t4 = ((const float4*)qrow)[i];
        q[4 * i + 0] = t4.x; q[4 * i + 1] = t4.y;
        q[4 * i + 2] = t4.z; q[4 * i + 3] = t4.w;
    }

    // ---- level 0: 64 unscaled scores q . k_rep ----
    for (int c = 0; c < 64; ++c) {
        const float4* kr = (const float4*)(ldsk + c * D_HEAD);
        float a = 0.f;
#pragma unroll
        for (int i = 0; i < 16; ++i) {
            const float4 kv = kr[i];
            a = fmaf(q[4 * i + 0], kv.x,
                fmaf(q[4 * i + 1], kv.y,
                fmaf(q[4 * i + 2], kv.z,
                fmaf(q[4 * i + 3], kv.w, a))));
        }
        scl0[c * PB + tid] = a;
    }

    // ---- level-0 top-8 (stable: descending value, lowest index first) ----
    {
        unsigned long long used = 0ull;
        int nb[8];
#pragma unroll
        for (int i = 0; i < 8; ++i) {
            float best = -3.402823466e38f; int bj = 0;
            for (int j = 0; j < 64; ++j) {
                const float v = scl0[j * PB + tid];
                const bool ok = !((used >> j) & 1ull) && (v > best);
                best = ok ? v : best;
                bj   = ok ? j : bj;
            }
            used |= 1ull << bj;
            nb[i] = bj * 32;                      // block base, bs = 32
        }
#pragma unroll
        for (int i = 0; i < 8; ++i) basel[i * PB + tid] = nb[i];
    }

    // ---- refinement levels: cs = 16, 8, 4, 2 ----
#pragma unroll
    for (int lev = 0; lev < 4; ++lev) {
        const int cs = 16 >> lev;
        for (int j = 0; j < 16; ++j) {
            const int rep = basel[(j >> 1) * PB + tid] + (j & 1) * cs + (cs >> 1);
            const float4* kr = (const float4*)(kbase + (size_t)rep * D_HEAD);
            float a = 0.f;
#pragma unroll
            for (int i = 0; i < 16; ++i) {
                const float4 kv = kr[i];
                a = fmaf(q[4 * i + 0], kv.x,
                    fmaf(q[4 * i + 1], kv.y,
                    fmaf(q[4 * i + 2], kv.z,
                    fmaf(q[4 * i + 3], kv.w, a))));
            }
            scl1[j * PB + tid] = a;
        }
        unsigned um = 0; int nb[8];
#pragma unroll
        for (int i = 0; i < 8; ++i) {
            float best = -3.402823466e38f; int bj = 0;
            for (int j = 0; j < 16; ++j) {
                const float v = scl1[j * PB + tid];
                const bool ok = !((um >> j) & 1u) && (v > best);
                best = ok ? v : best;
                bj   = ok ? j : bj;
            }
            um |= 1u << bj;
            nb[i] = basel[(bj >> 1) * PB + tid] + (bj & 1) * cs;
        }
#pragma unroll
        for (int i = 0; i < 8; ++i) basel[i * PB + tid] = nb[i];
    }

    // ---- final sparse attention over 16 keys (bases are pairs now) ----
    const float scale = 0.125f;                    // 1/sqrt(64)
    float mx = -3.402823466e38f;
    for (int j = 0; j < 16; ++j) {
        const int key = basel[(j >> 1) * PB + tid] + (j & 1);
        const float4* kr = (const float4*)(kbase + (size_t)key * D_HEAD);
        float a = 0.f;
#pragma unroll
        for (int i = 0; i < 16; ++i) {
            const float4 kv = kr[i];
            a = fmaf(q[4 * i + 0], kv.x,
                fmaf(q[4 * i + 1], kv.y,
                fmaf(q[4 * i + 2], kv.z,
                fmaf(q[4 * i + 3], kv.w, a))));
        }
        a *= scale;
        scl1[j * PB + tid] = a;
        mx = fmaxf(mx, a);
    }

    float ctx[D_HEAD];
#pragma unroll
    for (int d = 0; d < D_HEAD; ++d) ctx[d] = 0.f;
    float sum = 0.f;
    for (int j = 0; j < 16; ++j) {
        const float w = expf(scl1[j * PB + tid] - mx);
        sum += w;
        const int key = basel[(j >> 1) * PB + tid] + (j & 1);
        const float4* vr = (const float4*)(vbase + (size_t)key * D_HEAD);
#pragma unroll
        for (int i = 0; i < 16; ++i) {
            const float4 vv = vr[i];
            ctx[4 * i + 0] = fmaf(w, vv.x, ctx[4 * i + 0]);
            ctx[4 * i + 1] = fmaf(w, vv.y, ctx[4 * i + 1]);
            ctx[4 * i + 2] = fmaf(w, vv.z, ctx[4 * i + 2]);
            ctx[4 * i + 3] = fmaf(w, vv.w, ctx[4 * i + 3]);
        }
    }
    const float inv = 1.f / sum;

    // ctx -> [B,S,H,D] so the O-projection is a plain row-major GEMM
    const int b = bh >> 4, h = bh & (H_NUM - 1);
    float* __restrict__ orow =
        CTX + (((size_t)b * S_LEN + s) * H_NUM + h) * D_HEAD;
#pragma unroll
    for (int i = 0; i < 16; ++i) {
        float4 o;
        o.x = ctx[4 * i + 0] * inv; o.y = ctx[4 * i + 1] * inv;
        o.z = ctx[4 * i + 2] * inv; o.w = ctx[4 * i + 3] * inv;
        ((float4*)orow)[i] = o;
    }
}

// ---------------------------------------------------------------------------
extern "C" void kernel_launch(void* const* d_in, const int* in_sizes, int n_in,
                              void* d_out, int out_size, void* d_ws, size_t ws_size,
                              hipStream_t stream) {
    const float* x  = (const float*)d_in[0];
    const float* Wq = (const float*)d_in[1];
    const float* bq = (const float*)d_in[2];
    const float* Wk = (const float*)d_in[3];
    const float* bk = (const float*)d_in[4];
    const float* Wv = (const float*)d_in[5];
    const float* bv = (const float*)d_in[6];
    const float* Wo = (const float*)d_in[7];
    const float* bo = (const float*)d_in[8];
    float* out = (float*)d_out;

    const size_t PER = (size_t)2 * H_NUM * S_LEN * D_HEAD;  // 4 Mi floats
    float* qb = (float*)d_ws;
    float* kb = qb + PER;
    float* vb = kb + PER;
    float* cb = vb + PER;

    dim3 ggrid(32, 16, 1);   // M=4096 / 128, N=1024 / 64
    dim3 gblk(256, 1, 1);
    gemm_f32_wmma<<<ggrid, gblk, 0, stream>>>(x, Wq, bq, qb, 1);
    gemm_f32_wmma<<<ggrid, gblk, 0, stream>>>(x, Wk, bk, kb, 1);
    gemm_f32_wmma<<<ggrid, gblk, 0, stream>>>(x, Wv, bv, vb, 1);

    hip_prune_attn<<<dim3(65536 / PB), dim3(PB), 0, stream>>>(qb, kb, vb, cb);

    gemm_f32_wmma<<<ggrid, gblk, 0, stream>>>(cb, Wo, bo, out, 0);
}